// GMM_73933567033423
// MI455X (gfx1250) — compile-verified
//
#include <hip/hip_runtime.h>
#include <math.h>

typedef float v2f __attribute__((ext_vector_type(2)));
typedef float v8f __attribute__((ext_vector_type(8)));

#define KCOMP 4
#define DDIM  16

// ---------------------------------------------------------------------------
// Phase 1: per-component preprocessing (tiny, one wave).
//   Linv[k] = inverse of upper-triangular L[k]   (back substitution)
//   w[k]    = Linv[k] @ mu[k]
//   c[k]    = log(phi[k]) - 0.5*(D*log(2*pi) + 2*sum(log(diag(L[k]))))
// ---------------------------------------------------------------------------
__global__ void gmm_prep_kernel(const float* __restrict__ phi,
                                const float* __restrict__ mu,
                                const float* __restrict__ L,
                                float* __restrict__ wsLinv,   // [K][16][16]
                                float* __restrict__ wsW,      // [K][16]
                                float* __restrict__ wsC) {    // [K]
  int k = threadIdx.x;
  if (k >= KCOMP) return;
  const float* Lk = L + k * DDIM * DDIM;
  float X[DDIM][DDIM];
  for (int c = 0; c < DDIM; ++c) {
    for (int i = 0; i < DDIM; ++i) X[i][c] = 0.f;
    X[c][c] = 1.f / Lk[c * DDIM + c];
    for (int i = c - 1; i >= 0; --i) {
      float s = 0.f;
      for (int t = i + 1; t <= c; ++t) s += Lk[i * DDIM + t] * X[t][c];
      X[i][c] = -s / Lk[i * DDIM + i];
    }
  }
  for (int i = 0; i < DDIM; ++i)
    for (int j = 0; j < DDIM; ++j)
      wsLinv[(k * DDIM + i) * DDIM + j] = X[i][j];
  for (int i = 0; i < DDIM; ++i) {
    float s = 0.f;
    for (int j = 0; j < DDIM; ++j) s += X[i][j] * mu[k * DDIM + j];
    wsW[k * DDIM + i] = s;
  }
  float logdet = 0.f;
  for (int d = 0; d < DDIM; ++d) logdet += __logf(Lk[d * DDIM + d]);
  logdet *= 2.f;
  const float LOG2PI = 1.8378770664093453f;   // log(2*pi)
  wsC[k] = __logf(phi[k]) - 0.5f * ((float)DDIM * LOG2PI + logdet);
}

// ---------------------------------------------------------------------------
// Phase 2: main kernel. One wave32 owns one 16-point tile per loop iteration.
//   V(16x16) = Linv_k(16x16) * Z(16x16) - w_k  via 4x V_WMMA_F32_16X16X4_F32
//   maha[n]  = sum_d V[d][n]^2 (8 regs + cross-half shuffle)
//   out[n]   = -logsumexp_k( c_k - 0.5*maha_k[n] )
// ---------------------------------------------------------------------------
__global__ void __launch_bounds__(256) gmm_energy_kernel(
    const float* __restrict__ z, float* __restrict__ out,
    const float* __restrict__ wsLinv, const float* __restrict__ wsW,
    const float* __restrict__ wsC,
    int N, int numTiles, int numWaves) {
  const int lane  = threadIdx.x & 31;
  const int m     = lane & 15;    // A-row / B-column (point) index
  const int half  = lane >> 4;    // lane half selects K/M sub-blocks
  const int waveId = (int)((blockIdx.x * blockDim.x + threadIdx.x) >> 5);

  // A fragments: Linv rows in WMMA A layout (v2f per K-chunk), held in regs
  // across the whole tile loop.
  v2f a[KCOMP][4];
#pragma unroll
  for (int k = 0; k < KCOMP; ++k)
#pragma unroll
    for (int kk = 0; kk < 4; ++kk)
      a[k][kk] = *(const v2f*)(wsLinv + (k * 16 + m) * 16 + kk * 4 + 2 * half);

  // C-init fragments: -w_k in D-matrix layout (VGPR r holds row r + 8*half).
  v8f cinit[KCOMP];
#pragma unroll
  for (int k = 0; k < KCOMP; ++k) {
    const float* pw = wsW + k * 16 + 8 * half;
#pragma unroll
    for (int r = 0; r < 8; ++r) cinit[k][r] = -pw[r];
  }
  float ck[KCOMP];
#pragma unroll
  for (int k = 0; k < KCOMP; ++k) ck[k] = wsC[k];

  for (int t = waveId; t < numTiles; t += numWaves) {
    int n = t * 16 + m;
    if (n >= N) n = N - 1;                         // clamp tail (dup load ok)
    const float* zp = z + (size_t)n * 16 + 2 * half;
    // B fragments: each of the tile's 1KB of z loaded exactly once per wave.
    v2f b0 = *(const v2f*)(zp + 0);
    v2f b1 = *(const v2f*)(zp + 4);
    v2f b2 = *(const v2f*)(zp + 8);
    v2f b3 = *(const v2f*)(zp + 12);

    // Prefetch next tile (speculative; dropped if translation fails).
    int nt = t + numWaves; if (nt >= numTiles) nt = numTiles - 1;
    __builtin_prefetch(z + (size_t)nt * 256 + lane * 8, 0, 1);

    float logit[KCOMP];
#pragma unroll
    for (int k = 0; k < KCOMP; ++k) {
      v8f acc = cinit[k];
      acc = __builtin_amdgcn_wmma_f32_16x16x4_f32(false, a[k][0], false, b0,
                                                  (short)0, acc, false, false);
      acc = __builtin_amdgcn_wmma_f32_16x16x4_f32(false, a[k][1], false, b1,
                                                  (short)0, acc, false, false);
      acc = __builtin_amdgcn_wmma_f32_16x16x4_f32(false, a[k][2], false, b2,
                                                  (short)0, acc, false, false);
      acc = __builtin_amdgcn_wmma_f32_16x16x4_f32(false, a[k][3], false, b3,
                                                  (short)0, acc, false, false);
      float s = 0.f;
#pragma unroll
      for (int r = 0; r < 8; ++r) s += acc[r] * acc[r];
      s += __shfl_xor(s, 16, 32);                  // combine row halves
      logit[k] = ck[k] - 0.5f * s;
    }
    float mx = fmaxf(fmaxf(logit[0], logit[1]), fmaxf(logit[2], logit[3]));
    float se = __expf(logit[0] - mx) + __expf(logit[1] - mx) +
               __expf(logit[2] - mx) + __expf(logit[3] - mx);
    float e = -(mx + __logf(se));

    if (half == 0) {                                // lanes 0..15 own points
      int idx = t * 16 + m;
      if (idx < N) out[idx] = e;
    }
  }
}

// ---------------------------------------------------------------------------
// Launcher. Inputs (setup_inputs order): z[N*16], phi[4], mu[4*16], L[4*16*16]
// Output: energies[N] (float32). Workspace: 1092 floats.
// ---------------------------------------------------------------------------
extern "C" void kernel_launch(void* const* d_in, const int* in_sizes, int n_in,
                              void* d_out, int out_size, void* d_ws, size_t ws_size,
                              hipStream_t stream) {
  const float* z   = (const float*)d_in[0];
  const float* phi = (const float*)d_in[1];
  const float* mu  = (const float*)d_in[2];
  const float* L   = (const float*)d_in[3];
  float* out = (float*)d_out;
  int N = in_sizes[0] / DDIM;

  float* ws     = (float*)d_ws;
  float* wsLinv = ws;                 // 4*256 = 1024 floats
  float* wsW    = ws + 1024;          // 64 floats
  float* wsC    = ws + 1088;          // 4 floats

  gmm_prep_kernel<<<1, 32, 0, stream>>>(phi, mu, L, wsLinv, wsW, wsC);

  int numTiles = (N + 15) / 16;
  long wavesWanted = (numTiles + 7) / 8;           // ~8 tiles per wave
  const int wavesPerBlock = 8;                     // 256 threads
  long blocks = (wavesWanted + wavesPerBlock - 1) / wavesPerBlock;
  if (blocks < 1) blocks = 1;
  if (blocks > 16384) blocks = 16384;
  int numWaves = (int)(blocks * wavesPerBlock);

  gmm_energy_kernel<<<(int)blocks, 256, 0, stream>>>(z, out, wsLinv, wsW, wsC,
                                                     N, numTiles, numWaves);
}